// DGCNN_11106785427638
// MI455X (gfx1250) — compile-verified
//
#include <hip/hip_runtime.h>
#include <math.h>

typedef __attribute__((ext_vector_type(2))) float v2f;
typedef __attribute__((ext_vector_type(8))) float v8f;

// D = A(16x4) * B(4x16) + C, full f32 on the matrix pipe.
#define WMMA4(a, b, c) \
  __builtin_amdgcn_wmma_f32_16x16x4_f32(false, (a), false, (b), (short)0, (c), false, false)

static constexpr int BB  = 8;
static constexpr int NN  = 2048;
static constexpr int KK  = 20;
static constexpr int CLS = 40;
#define NEG_INF (-3.4e38f)

// Branchless guarded load: p[c*stride + off] if c < CAP else 0.
// When CAP is a multiple of 4 the padded K-loop never exceeds CAP -> plain load.
template <int CAP>
__device__ __forceinline__ float gclamp(const float* __restrict__ p, int stride,
                                        int c, int off) {
  if constexpr ((CAP & 3) == 0) {
    return p[(size_t)c * stride + off];
  } else {
    float v = p[(size_t)(c < CAP ? c : 0) * stride + off];
    return (c < CAP) ? v : 0.f;
  }
}

// ---------------------------------------------------------------------------
// 1) per-point squared norms: xx[b][n] = sum_c x[b][c][n]^2
// ---------------------------------------------------------------------------
template <int C>
__global__ __launch_bounds__(256) void k_sumsq(const float* __restrict__ xin,
                                               float* __restrict__ xx) {
  int i = blockIdx.x * 256 + threadIdx.x;   // over B*NN
  int b = i / NN, n = i - b * NN;
  const float* xb = xin + (size_t)b * C * NN;
  float s = 0.f;
#pragma unroll
  for (int c = 0; c < C; ++c) { float v = xb[(size_t)c * NN + n]; s += v * v; }
  xx[i] = s;
}

// ---------------------------------------------------------------------------
// 2) kNN: one workgroup = (b, 16-row tile). Phase A: WMMA f32 Gram tiles into
//    a 16 x 2048 LDS distance strip. Phase B: 20 iterative argmax passes,
//    16 threads per row, first-index tie-break (matches lax.top_k).
// ---------------------------------------------------------------------------
template <int C>
__global__ __launch_bounds__(256) void k_knn(const float* __restrict__ xin,
                                             const float* __restrict__ xx,
                                             int* __restrict__ idx) {
  constexpr int KSTEPS = (C + 3) >> 2;
  extern __shared__ float lds[];
  float* dist  = lds;                   // 16 * NN floats
  float* bestv = lds + 16 * NN;         // 256
  int*   besti = (int*)(bestv + 256);   // 256

  const int tid  = threadIdx.x;
  const int wave = tid >> 5;
  const int lane = tid & 31;
  const int r    = lane & 15;
  const int hh   = lane >> 4;
  const int b    = blockIdx.y;
  const int row0 = blockIdx.x * 16;
  const float* xb  = xin + (size_t)b * C * NN;
  const float* xxb = xx + (size_t)b * NN;

  v2f afrag[KSTEPS];
#pragma unroll
  for (int s = 0; s < KSTEPS; ++s) {
    int c0 = s * 4 + 2 * hh;
    v2f a;
    a.x = gclamp<C>(xb, NN, c0,     row0 + r);
    a.y = gclamp<C>(xb, NN, c0 + 1, row0 + r);
    afrag[s] = a;
  }
  float xxr[8];
#pragma unroll
  for (int v = 0; v < 8; ++v) xxr[v] = xxb[row0 + v + 8 * hh];

  for (int ct = wave; ct < NN / 16; ct += 8) {   // uniform per wave: EXEC all 1s
    int col0 = ct * 16;
    v8f acc = {};
#pragma unroll
    for (int s = 0; s < KSTEPS; ++s) {
      int c0 = s * 4 + 2 * hh;
      v2f bf;
      bf.x = gclamp<C>(xb, NN, c0,     col0 + r);
      bf.y = gclamp<C>(xb, NN, c0 + 1, col0 + r);
      acc = WMMA4(afrag[s], bf, acc);
    }
    float xxc = xxb[col0 + r];
#pragma unroll
    for (int v = 0; v < 8; ++v) {
      int m = v + 8 * hh;                               // D layout: M = v + 8h, N = lane%16
      dist[m * NN + col0 + r] = 2.f * acc[v] - xxr[v] - xxc;
    }
  }
  __syncthreads();

  const int rrow = tid >> 4, l16 = tid & 15;
  float* drow = dist + rrow * NN;
  for (int t = 0; t < KK; ++t) {
    float bv = NEG_INF;
    int   bi = 0x7fffffff;
    for (int j = l16; j < NN; j += 16) {
      float v = drow[j];
      if (v > bv) { bv = v; bi = j; }                   // keeps smallest index on ties
    }
    bestv[tid] = bv; besti[tid] = bi;
    __syncthreads();
    if (l16 == 0) {
      float wv = bestv[tid]; int wi = besti[tid];
      for (int q = 1; q < 16; ++q) {
        float v2 = bestv[tid + q]; int i2 = besti[tid + q];
        if (v2 > wv || (v2 == wv && i2 < wi)) { wv = v2; wi = i2; }
      }
      idx[((size_t)b * NN + row0 + rrow) * KK + t] = wi;
      drow[wi] = NEG_INF;
    }
    __syncthreads();
  }
}

// ---------------------------------------------------------------------------
// 3) EdgeConv (fused): gather edge features for 8 n's x 20 k's = 160 cols,
//    WMMA GEMM (64 x 160, K=2C) + BN + LeakyReLU(0.2) -> ytile in LDS,
//    then WMMA GEMM with w5[:, cofs:cofs+64] accumulated into h5 (B,64,N,K),
//    then k-max-pool -> xout (B,64,N).
//    Tile map: wave keeps a fixed M-tile (mt = wave&3) so A-frags (weight rows)
//    are preloaded once per wave and reused across 5 N-tiles.
// ---------------------------------------------------------------------------
template <int C>
__global__ __launch_bounds__(256) void k_edge(const float* __restrict__ xin,
                                              const int* __restrict__ idx,
                                              const float* __restrict__ W,
                                              const float* __restrict__ bnp,
                                              const float* __restrict__ w5, int cofs,
                                              float* __restrict__ h5,
                                              float* __restrict__ xout,
                                              int accum) {
  constexpr int C2 = 2 * C;
  constexpr int KP = ((C2 + 3) >> 2) << 2;   // padded K-dim (6->8, 128->128)
  constexpr int KSTEPS = KP >> 2;
  extern __shared__ float lds[];
  float* feat  = lds;                   // KP * 160
  float* ytile = lds + KP * 160;        // 64 * 160
  float* bnA   = ytile + 64 * 160;      // 64
  float* bnB   = bnA + 64;              // 64
  int*   idxl  = (int*)(bnB + 64);      // 160

  const int tid  = threadIdx.x;
  const int wave = tid >> 5;
  const int lane = tid & 31;
  const int r    = lane & 15;
  const int hh   = lane >> 4;
  const int b    = blockIdx.y;
  const int n0   = blockIdx.x * 8;
  const float* xb = xin + (size_t)b * C * NN;

  const int mt     = wave & 3;        // fixed M-tile per wave
  const int nthalf = wave >> 2;       // 0/1 -> N-tiles [0..4] or [5..9]
  const int orow   = mt * 16 + r;

  // Preload GEMM1 A-frags (weight rows) once per wave.
  v2f afA[KSTEPS];
#pragma unroll
  for (int s = 0; s < KSTEPS; ++s) {
    int c0 = s * 4 + 2 * hh;
    v2f a;
    a.x = gclamp<C2>(W + orow * C2, 1, c0, 0);
    a.y = gclamp<C2>(W + orow * C2, 1, c0 + 1, 0);
    afA[s] = a;
  }

  if (tid < 160) {
    int ln = tid / 20, kk2 = tid - ln * 20;
    idxl[tid] = idx[((size_t)b * NN + n0 + ln) * KK + kk2];
  }
  if (tid < 64) {
    float s = bnp[tid], bb = bnp[64 + tid], mm = bnp[128 + tid], vv = bnp[192 + tid];
    float A = s * rsqrtf(vv + 1e-5f);
    bnA[tid] = A;
    bnB[tid] = bb - mm * A;
  }
  __syncthreads();

  for (int e = tid; e < KP * 160; e += 256) {
    int c = e / 160, p = e - c * 160;
    int ln = p / 20;
    int n = n0 + ln;
    float v = 0.f;
    if (c < C) {
      int j = idxl[p];
      v = xb[(size_t)c * NN + j] - xb[(size_t)c * NN + n];
    } else if (c < C2) {
      v = xb[(size_t)(c - C) * NN + n];
    }
    feat[e] = v;
  }
  __syncthreads();

  // GEMM1: ytile = lrelu(bn(W @ feat))
#pragma unroll
  for (int tl = 0; tl < 5; ++tl) {
    int nt  = nthalf * 5 + tl;
    int col = nt * 16 + r;
    v8f acc = {};
#pragma unroll
    for (int s = 0; s < KSTEPS; ++s) {
      int c0 = s * 4 + 2 * hh;
      v2f bf = { feat[c0 * 160 + col], feat[(c0 + 1) * 160 + col] };
      acc = WMMA4(afA[s], bf, acc);
    }
#pragma unroll
    for (int v = 0; v < 8; ++v) {
      int m = mt * 16 + v + 8 * hh;
      float y = acc[v] * bnA[m] + bnB[m];
      y = (y >= 0.f) ? y : 0.2f * y;
      ytile[m * 160 + col] = y;
    }
  }

  // Preload GEMM2 A-frags: w5[:, cofs:cofs+64] rows (w5 is 64 x 256 row-major).
  v2f afB[16];
#pragma unroll
  for (int s = 0; s < 16; ++s) {
    int c0 = s * 4 + 2 * hh;
    v2f a = { w5[orow * 256 + cofs + c0], w5[orow * 256 + cofs + c0 + 1] };
    afB[s] = a;
  }
  __syncthreads();

  // GEMM2: h5 += w5_blk @ ytile
#pragma unroll
  for (int tl = 0; tl < 5; ++tl) {
    int nt  = nthalf * 5 + tl;
    int col = nt * 16 + r;
    v8f acc = {};
#pragma unroll
    for (int s = 0; s < 16; ++s) {
      int c0 = s * 4 + 2 * hh;
      v2f bf = { ytile[c0 * 160 + col], ytile[(c0 + 1) * 160 + col] };
      acc = WMMA4(afB[s], bf, acc);
    }
    int ln = col / 20, kk2 = col - ln * 20;
    int n = n0 + ln;
#pragma unroll
    for (int v = 0; v < 8; ++v) {
      int m = mt * 16 + v + 8 * hh;
      size_t off = (((size_t)b * 64 + m) * NN + n) * KK + kk2;
      float val = acc[v];
      if (accum) val += h5[off];
      h5[off] = val;
    }
  }

  if (xout) {
    for (int e = tid; e < 512; e += 256) {    // 64 channels x 8 n's
      int m = e >> 3, ln = e & 7;
      float mx = NEG_INF;
#pragma unroll
      for (int kk2 = 0; kk2 < KK; ++kk2) mx = fmaxf(mx, ytile[m * 160 + ln * 20 + kk2]);
      xout[((size_t)b * 64 + m) * NN + n0 + ln] = mx;
    }
  }
}

// ---------------------------------------------------------------------------
// 4) stage5: h = max_k( lrelu( bn5( h5 ) ) ), h5 layout (B,64,N,K)
// ---------------------------------------------------------------------------
__global__ __launch_bounds__(256) void k_stage5(const float* __restrict__ h5,
                                                const float* __restrict__ bnp,
                                                float* __restrict__ h) {
  int i = blockIdx.x * 256 + threadIdx.x;   // over B*64*NN
  int c = (i / NN) & 63;
  float A  = bnp[c] * rsqrtf(bnp[192 + c] + 1e-5f);
  float Bc = bnp[64 + c] - bnp[128 + c] * A;
  const float* p = h5 + (size_t)i * KK;
  float mx = NEG_INF;
#pragma unroll
  for (int kk2 = 0; kk2 < KK; ++kk2) {
    float y = p[kk2] * A + Bc;
    y = (y >= 0.f) ? y : 0.2f * y;
    mx = fmaxf(mx, y);
  }
  h[i] = mx;
}

// ---------------------------------------------------------------------------
// 5) stage6: h2 = lrelu( bn6( w6 @ h ) ), w6 is 256x64, WMMA GEMM per 64-col
//    tile. Each wave owns 2 M-tiles; A-frags preloaded per M-tile, reused
//    across 4 N-tiles.
// ---------------------------------------------------------------------------
__global__ __launch_bounds__(256) void k_stage6(const float* __restrict__ h,
                                                const float* __restrict__ w6,
                                                const float* __restrict__ bnp,
                                                float* __restrict__ h2) {
  __shared__ float htile[64 * 64];
  __shared__ float bnA[256], bnB[256];
  const int tid  = threadIdx.x;
  const int wave = tid >> 5, lane = tid & 31;
  const int r = lane & 15, hh = lane >> 4;
  const int b  = blockIdx.y;
  const int n0 = blockIdx.x * 64;

  for (int e = tid; e < 64 * 64; e += 256) {
    int c = e >> 6, j = e & 63;
    htile[e] = h[((size_t)b * 64 + c) * NN + n0 + j];
  }
  {
    float s = bnp[tid], bb = bnp[256 + tid], mm = bnp[512 + tid], vv = bnp[768 + tid];
    float A = s * rsqrtf(vv + 1e-5f);
    bnA[tid] = A;
    bnB[tid] = bb - mm * A;
  }
  __syncthreads();

#pragma unroll
  for (int half = 0; half < 2; ++half) {
    int mt = wave * 2 + half;
    int orow = mt * 16 + r;
    v2f af[16];
#pragma unroll
    for (int s = 0; s < 16; ++s) {
      int c0 = s * 4 + 2 * hh;
      v2f a = { w6[orow * 64 + c0], w6[orow * 64 + c0 + 1] };
      af[s] = a;
    }
#pragma unroll
    for (int nt = 0; nt < 4; ++nt) {
      int col = nt * 16 + r;
      v8f acc = {};
#pragma unroll
      for (int s = 0; s < 16; ++s) {
        int c0 = s * 4 + 2 * hh;
        v2f bf = { htile[c0 * 64 + col], htile[(c0 + 1) * 64 + col] };
        acc = WMMA4(af[s], bf, acc);
      }
#pragma unroll
      for (int v = 0; v < 8; ++v) {
        int m = mt * 16 + v + 8 * hh;
        float y = acc[v] * bnA[m] + bnB[m];
        y = (y >= 0.f) ? y : 0.2f * y;
        h2[((size_t)b * 256 + m) * NN + n0 + col] = y;
      }
    }
  }
}

// ---------------------------------------------------------------------------
// 6) global max + mean pooling over N: g (B,512) = [max(h2), mean(h2)]
// ---------------------------------------------------------------------------
__global__ __launch_bounds__(256) void k_pool(const float* __restrict__ h2,
                                              float* __restrict__ g) {
  int b = blockIdx.x, c = threadIdx.x;   // 256 channels
  const float* p = h2 + ((size_t)b * 256 + c) * NN;
  float mx = NEG_INF, sm = 0.f;
  for (int n = 0; n < NN; ++n) { float v = p[n]; mx = fmaxf(mx, v); sm += v; }
  g[b * 512 + c] = mx;
  g[b * 512 + 256 + c] = sm * (1.0f / NN);
}

// ---------------------------------------------------------------------------
// 7) head MLP: 512 -> 256 -> 64 -> CLS, with BN + lrelu(0.01), one block
// ---------------------------------------------------------------------------
__global__ __launch_bounds__(256) void k_head(const float* __restrict__ g,
                                              const float* __restrict__ lw1, const float* __restrict__ lb1,
                                              const float* __restrict__ lbn1,
                                              const float* __restrict__ lw2, const float* __restrict__ lb2,
                                              const float* __restrict__ lbn2,
                                              const float* __restrict__ lw3, const float* __restrict__ lb3,
                                              float* __restrict__ out) {
  __shared__ float gb[8 * 512];
  __shared__ float z1[8 * 256];
  __shared__ float z2[8 * 64];
  int tid = threadIdx.x;
  for (int e = tid; e < 8 * 512; e += 256) gb[e] = g[e];
  __syncthreads();
  for (int e = tid; e < 8 * 256; e += 256) {
    int row = e >> 8, col = e & 255;
    float s = 0.f;
    for (int j = 0; j < 512; ++j) s += gb[row * 512 + j] * lw1[col * 512 + j];
    s += lb1[col];
    float A = lbn1[col] * rsqrtf(lbn1[768 + col] + 1e-5f);
    s = (s - lbn1[512 + col]) * A + lbn1[256 + col];
    z1[e] = (s >= 0.f) ? s : 0.01f * s;
  }
  __syncthreads();
  for (int e = tid; e < 8 * 64; e += 256) {
    int row = e >> 6, col = e & 63;
    float s = 0.f;
    for (int j = 0; j < 256; ++j) s += z1[row * 256 + j] * lw2[col * 256 + j];
    s += lb2[col];
    float A = lbn2[col] * rsqrtf(lbn2[192 + col] + 1e-5f);
    s = (s - lbn2[128 + col]) * A + lbn2[64 + col];
    z2[e] = (s >= 0.f) ? s : 0.01f * s;
  }
  __syncthreads();
  for (int e = tid; e < 8 * CLS; e += 256) {
    int row = e / CLS, col = e - row * CLS;
    float s = 0.f;
    for (int j = 0; j < 64; ++j) s += z2[row * 64 + j] * lw3[col * 64 + j];
    out[e] = s + lb3[col];
  }
}

// ---------------------------------------------------------------------------
template <int C>
static void run_block(const float* xin, const float* W, const float* bnp, int cofs,
                      float* xout, int accum, float* xxbuf, int* idxbf, float* h5buf,
                      const float* w5, hipStream_t stream) {
  constexpr int KP = ((2 * C + 3) >> 2) << 2;
  const size_t knn_lds = (size_t)(16 * NN + 512) * 4;
  const size_t elds = (size_t)(KP * 160 + 64 * 160 + 128 + 160) * 4;
  k_sumsq<C><<<(BB * NN) / 256, 256, 0, stream>>>(xin, xxbuf);
  k_knn<C><<<dim3(NN / 16, BB), 256, knn_lds, stream>>>(xin, xxbuf, idxbf);
  k_edge<C><<<dim3(NN / 8, BB), 256, elds, stream>>>(xin, idxbf, W, bnp, w5, cofs,
                                                     h5buf, xout, accum);
}

extern "C" void kernel_launch(void* const* d_in, const int* in_sizes, int n_in,
                              void* d_out, int out_size, void* d_ws, size_t ws_size,
                              hipStream_t stream) {
  (void)in_sizes; (void)n_in; (void)out_size; (void)ws_size;

  const float* x    = (const float*)d_in[0];
  const float* w1   = (const float*)d_in[1];
  const float* bn1  = (const float*)d_in[2];
  const float* w2   = (const float*)d_in[3];
  const float* bn2  = (const float*)d_in[4];
  const float* w3   = (const float*)d_in[5];
  const float* bn3  = (const float*)d_in[6];
  const float* w4   = (const float*)d_in[7];
  const float* bn4  = (const float*)d_in[8];
  const float* w5   = (const float*)d_in[9];
  const float* bn5  = (const float*)d_in[10];
  const float* w6   = (const float*)d_in[11];
  const float* bn6  = (const float*)d_in[12];
  const float* lw1  = (const float*)d_in[13];
  const float* lb1  = (const float*)d_in[14];
  const float* lbn1 = (const float*)d_in[15];
  const float* lw2  = (const float*)d_in[16];
  const float* lb2  = (const float*)d_in[17];
  const float* lbn2 = (const float*)d_in[18];
  const float* lw3  = (const float*)d_in[19];
  const float* lb3  = (const float*)d_in[20];

  char* wsb = (char*)d_ws;
  size_t off = 0;
  auto take = [&](size_t bytes) -> char* {
    char* p = wsb + off;
    off = (off + bytes + 255) & ~(size_t)255;
    return p;
  };
  float* xxbuf = (float*)take((size_t)BB * NN * 4);
  int*   idxbf = (int*)  take((size_t)BB * NN * KK * 4);
  float* h5buf = (float*)take((size_t)BB * 64 * NN * KK * 4);
  float* x1buf = (float*)take((size_t)BB * 64 * NN * 4);
  float* x2buf = (float*)take((size_t)BB * 64 * NN * 4);
  float* x3buf = (float*)take((size_t)BB * 64 * NN * 4);
  float* hbuf  = (float*)take((size_t)BB * 64 * NN * 4);
  float* h2buf = (float*)take((size_t)BB * 256 * NN * 4);
  float* gbuf  = (float*)take((size_t)BB * 512 * 4);

  run_block<3> (x,     w1, bn1, 0,   x1buf, 0, xxbuf, idxbf, h5buf, w5, stream);
  run_block<64>(x1buf, w2, bn2, 64,  x2buf, 1, xxbuf, idxbf, h5buf, w5, stream);
  run_block<64>(x2buf, w3, bn3, 128, x3buf, 1, xxbuf, idxbf, h5buf, w5, stream);
  run_block<64>(x3buf, w4, bn4, 192, nullptr, 1, xxbuf, idxbf, h5buf, w5, stream);

  k_stage5<<<(BB * 64 * NN) / 256, 256, 0, stream>>>(h5buf, bn5, hbuf);
  k_stage6<<<dim3(NN / 64, BB), 256, 0, stream>>>(hbuf, w6, bn6, h2buf);
  k_pool<<<BB, 256, 0, stream>>>(h2buf, gbuf);
  k_head<<<1, 256, 0, stream>>>(gbuf, lw1, lb1, lbn1, lw2, lb2, lbn2, lw3, lb3,
                                (float*)d_out);
}